// LateInteractionScorer_18262200943122
// MI455X (gfx1250) — compile-verified
//
#include <hip/hip_runtime.h>
#include <cstdint>

typedef __attribute__((ext_vector_type(2))) float v2f;
typedef __attribute__((ext_vector_type(8))) float v8f;

#define BATCH            64
#define QLEN             32
#define DLEN             4096
#define HD               128
#define KSTEPS           (HD / 4)                 // 32 WMMA K-steps
#define QSTRIDE          132                      // 128 + 4 pad floats -> conflict-free LDS reads
#define BLOCKS_PER_BATCH 8
#define DOCS_PER_BLOCK   (DLEN / BLOCKS_PER_BATCH)           // 512
#define WAVES_PER_BLOCK  8
#define DOCS_PER_WAVE    (DOCS_PER_BLOCK / WAVES_PER_BLOCK)  // 64
#define TILES_PER_WAVE   (DOCS_PER_WAVE / 16)                // 4
#define NEGBIG           (-1.0e30f)

__device__ __forceinline__ void atomicMaxF32(float* addr, float val) {
  // Standard bit-trick float max: order-independent, deterministic.
  if (val >= 0.0f)
    atomicMax((int*)addr, __float_as_int(val));
  else
    atomicMin((unsigned int*)addr, __float_as_uint(val));
}

__global__ void lis_init(float* __restrict__ ws) {
  int i = blockIdx.x * blockDim.x + threadIdx.x;
  if (i < BATCH * QLEN) ws[i] = -__builtin_inff();
}

__global__ void __launch_bounds__(256)
lis_main(const float* __restrict__ Q, const float* __restrict__ D,
         const int* __restrict__ Mk, float* __restrict__ ws) {
  __shared__ float qs[QLEN * QSTRIDE];   // 16.9 KB, padded rows

  const int b   = blockIdx.y;
  const int tid = threadIdx.x;

  // ---- Cooperative load of Q[b] (32x128 f32) into padded LDS, float4 wide ----
  const float4* qg = (const float4*)(Q + (size_t)b * QLEN * HD);
  #pragma unroll
  for (int it = 0; it < (QLEN * HD / 4) / 256; ++it) {   // 1024 float4 / 256 thr = 4 iters
    int idx = it * 256 + tid;
    int row = idx >> 5;        // 32 float4 per 128-float row
    int c4  = idx & 31;
    float4 v = qg[idx];
    *(float4*)&qs[row * QSTRIDE + c4 * 4] = v;           // 16B aligned (528 % 16 == 0)
  }
  __syncthreads();

  const int lane = tid & 31;
  const int wv   = tid >> 5;
  const int lo   = lane & 15;      // doc column / q row within tile
  const int hi   = lane >> 4;      // selects K pair {2hi, 2hi+1}

  const float* qrow0 = qs + lo * QSTRIDE + 2 * hi;       // A, q-tile 0 (q = lo)
  const float* qrow1 = qrow0 + 16 * QSTRIDE;             // A, q-tile 1 (q = 16 + lo)

  const size_t dBatch = (size_t)b * DLEN * HD;
  const int    dblock = blockIdx.x * DOCS_PER_BLOCK + wv * DOCS_PER_WAVE;
  // per-lane base within a tile: row (dbase+lo), cols {2hi, 2hi+1} + 4k
  const float* dwave  = D + dBatch + (size_t)lo * HD + 2 * hi;

  v8f mx0, mx1;
  #pragma unroll
  for (int i = 0; i < 8; ++i) { mx0[i] = -__builtin_inff(); mx1[i] = -__builtin_inff(); }

  #pragma unroll
  for (int t = 0; t < TILES_PER_WAVE; ++t) {
    const int    dbase = dblock + t * 16;
    const float* drow  = dwave + (size_t)dbase * HD;

    if (t + 1 < TILES_PER_WAVE)  // warm L2/WGP$ for the next 16-doc tile
      __builtin_prefetch(dwave + (size_t)(dbase + 16) * HD, 0, 3);

    // ---- Stage the whole B tile (32 independent global_load_b64) + mask ----
    v2f breg[KSTEPS];
    #pragma unroll
    for (int kk = 0; kk < KSTEPS; ++kk)
      breg[kk] = *(const v2f*)(drow + 4 * kk);
    const int mval = Mk[(size_t)b * DLEN + dbase + lo];   // issued with the batch

    v8f c0 = {0.f,0.f,0.f,0.f,0.f,0.f,0.f,0.f};
    v8f c1 = {0.f,0.f,0.f,0.f,0.f,0.f,0.f,0.f};

    // ---- 64 WMMAs fed from registers (B) and LDS (A) ----
    #pragma unroll
    for (int kk = 0; kk < KSTEPS; ++kk) {
      v2f a0 = *(const v2f*)(qrow0 + 4 * kk);
      v2f a1 = *(const v2f*)(qrow1 + 4 * kk);
      c0 = __builtin_amdgcn_wmma_f32_16x16x4_f32(false, a0, false, breg[kk],
                                                 (short)0, c0, false, false);
      c1 = __builtin_amdgcn_wmma_f32_16x16x4_f32(false, a1, false, breg[kk],
                                                 (short)0, c1, false, false);
    }

    // mask penalty depends only on this lane's doc column d = dbase + lo
    const float pen = mval ? 0.0f : NEGBIG;
    #pragma unroll
    for (int i = 0; i < 8; ++i) {
      mx0[i] = fmaxf(mx0[i], c0[i] + pen);
      mx1[i] = fmaxf(mx1[i], c1[i] + pen);
    }
  }

  // ---- Reduce max over the 16 doc-lanes that share each q, then atomic-combine ----
  #pragma unroll
  for (int i = 0; i < 8; ++i) {
    float m0 = mx0[i], m1 = mx1[i];
    #pragma unroll
    for (int off = 8; off >= 1; off >>= 1) {   // xor < 16 stays within each 16-lane half
      m0 = fmaxf(m0, __shfl_xor(m0, off, 32));
      m1 = fmaxf(m1, __shfl_xor(m1, off, 32));
    }
    if (lo == 0) {
      const int qb = hi ? (i + 8) : i;         // C layout: q = vgpr + 8*(lane/16)
      atomicMaxF32(&ws[b * QLEN + qb],      m0);   // q-tile 0 -> q in [0,16)
      atomicMaxF32(&ws[b * QLEN + 16 + qb], m1);   // q-tile 1 -> q in [16,32)
    }
  }
}

__global__ void lis_reduce(const float* __restrict__ ws, float* __restrict__ out) {
  const int b = blockIdx.x;
  const int q = threadIdx.x;                   // 32 threads = 1 wave
  float v = ws[b * QLEN + q];
  #pragma unroll
  for (int off = 16; off >= 1; off >>= 1)
    v += __shfl_xor(v, off, 32);
  if (q == 0) out[b] = v;
}

extern "C" void kernel_launch(void* const* d_in, const int* in_sizes, int n_in,
                              void* d_out, int out_size, void* d_ws, size_t ws_size,
                              hipStream_t stream) {
  const float* Q   = (const float*)d_in[0];   // [64, 32, 128] f32
  const float* D   = (const float*)d_in[1];   // [64, 4096, 128] f32
  const int*   Mk  = (const int*)d_in[2];     // [64, 4096] i32
  float*       out = (float*)d_out;           // [64] f32
  float*       ws  = (float*)d_ws;            // needs 64*32*4 = 8 KB

  lis_init  <<<dim3((BATCH * QLEN + 255) / 256), dim3(256), 0, stream>>>(ws);
  lis_main  <<<dim3(BLOCKS_PER_BATCH, BATCH),    dim3(256), 0, stream>>>(Q, D, Mk, ws);
  lis_reduce<<<dim3(BATCH),                      dim3(32),  0, stream>>>(ws, out);
}